// GCN_67765993996385
// MI455X (gfx1250) — compile-verified
//
#include <hip/hip_runtime.h>
#include <hip/hip_bf16.h>

typedef float v2f __attribute__((ext_vector_type(2)));
typedef float v8f __attribute__((ext_vector_type(8)));

#define GCN_N 50000
#define GCN_E 800000
#define GCN_D 64

// ---------------------------------------------------------------------------
// Degree / normalization: deg[i] = (#edges with dst==i) + 1 (self loop),
// dinv[i] = rsqrt(deg[i]).  deg accumulated in-place in the dinv buffer.
// ---------------------------------------------------------------------------
__global__ void k_zero_f32(float* __restrict__ p, int n) {
    int i = blockIdx.x * blockDim.x + threadIdx.x;
    if (i < n) p[i] = 0.0f;
}

__global__ void k_count_deg(const long long* __restrict__ ei,
                            float* __restrict__ deg, int E) {
    int e = blockIdx.x * blockDim.x + threadIdx.x;
    if (e < E) {
        int d = (int)ei[E + e];          // dst row of edge_index
        atomicAdd(&deg[d], 1.0f);
    }
}

__global__ void k_deg_to_dinv(float* __restrict__ deg, int n) {
    int i = blockIdx.x * blockDim.x + threadIdx.x;
    if (i < n) {
        float dg = deg[i] + 1.0f;        // + self loop
        deg[i] = rsqrtf(dg);             // deg >= 1 always
    }
}

// ---------------------------------------------------------------------------
// H = X @ W  via V_WMMA_F32_16X16X4_F32.  One wave per 16-row tile, covering
// all 64 output columns with 4 accumulators (A fragment reused 4x per K step).
// N is a multiple of 16 (50000 = 3125*16) so every tile is full and EXEC
// stays all-ones across the WMMA ops, as the ISA requires.
// A layout (16x4 f32): lane 0-15 -> M=lane, VGPR0/1 = K={0,1}; lane 16-31 ->
// M=lane-16, K={2,3}.  C/D layout: VGPR r -> row r (+8 for upper half-wave),
// col = lane&15.
// ---------------------------------------------------------------------------
__global__ void k_gemm64_wmma(const float* __restrict__ X,
                              const float* __restrict__ W,
                              float* __restrict__ H, int nRowTiles) {
    const int wavesPerBlock = blockDim.x >> 5;
    const int rowTile = blockIdx.x * wavesPerBlock + (threadIdx.x >> 5);
    if (rowTile >= nRowTiles) return;    // wave-uniform exit

    const int lane = threadIdx.x & 31;
    const int m  = lane & 15;            // row within tile (A), col within tile (B/C)
    const int kh = lane >> 4;            // K half-select for A fragment

    const float* __restrict__ xr = X + (rowTile * 16 + m) * GCN_D;

    v8f acc0 = {}, acc1 = {}, acc2 = {}, acc3 = {};

    for (int kk = 0; kk < GCN_D; kk += 4) {
        const int ka = kk + 2 * kh;
        v2f a;
        a.x = xr[ka];
        a.y = xr[ka + 1];

        const float* __restrict__ w0 = W + ka * GCN_D;        // row ka of W
        const float* __restrict__ w1 = W + (ka + 1) * GCN_D;  // row ka+1 of W
        v2f b0, b1c, b2, b3;
        b0.x  = w0[m];       b0.y  = w1[m];
        b1c.x = w0[16 + m];  b1c.y = w1[16 + m];
        b2.x  = w0[32 + m];  b2.y  = w1[32 + m];
        b3.x  = w0[48 + m];  b3.y  = w1[48 + m];

        acc0 = __builtin_amdgcn_wmma_f32_16x16x4_f32(false, a, false, b0,  (short)0, acc0, false, false);
        acc1 = __builtin_amdgcn_wmma_f32_16x16x4_f32(false, a, false, b1c, (short)0, acc1, false, false);
        acc2 = __builtin_amdgcn_wmma_f32_16x16x4_f32(false, a, false, b2,  (short)0, acc2, false, false);
        acc3 = __builtin_amdgcn_wmma_f32_16x16x4_f32(false, a, false, b3,  (short)0, acc3, false, false);
    }

#pragma unroll
    for (int r = 0; r < 8; ++r) {
        const int orow = rowTile * 16 + r + 8 * kh;
        float* __restrict__ o = H + orow * GCN_D + m;
        o[0]  = acc0[r];
        o[16] = acc1[r];
        o[32] = acc2[r];
        o[48] = acc3[r];
    }
}

// ---------------------------------------------------------------------------
// agg = dinv[i]^2 * h[i]   (self-loop contribution, also initializes agg)
// One thread per float4.
// ---------------------------------------------------------------------------
__global__ void k_self_init(const float* __restrict__ h,
                            const float* __restrict__ dinv,
                            float* __restrict__ agg, int nQuads) {
    int t = blockIdx.x * blockDim.x + threadIdx.x;
    if (t < nQuads) {
        int node = t >> 4;               // 16 float4 per node (D=64)
        float w = dinv[node];
        w = w * w;
        float4 hv = ((const float4*)h)[t];
        float4 o;
        o.x = hv.x * w; o.y = hv.y * w; o.z = hv.z * w; o.w = hv.w * w;
        ((float4*)agg)[t] = o;
    }
}

// ---------------------------------------------------------------------------
// Edge scatter: agg[dst] += dinv[src]*dinv[dst] * h[src].
// 16 lanes per edge, float4 (B128) gather per lane, f32 global atomics for
// the scatter.  h and agg are 12.8 MB each -> resident in the 192 MB L2, so
// both the gathers and the atomic RMWs are served near-memory in L2.
// ---------------------------------------------------------------------------
__global__ void k_edge_scatter(const long long* __restrict__ ei,
                               const float* __restrict__ h,
                               const float* __restrict__ dinv,
                               float* __restrict__ agg, int E) {
    int t = blockIdx.x * blockDim.x + threadIdx.x;
    int e = t >> 4;
    if (e >= E) return;
    int f = (t & 15) << 2;
    int s = (int)ei[e];
    int d = (int)ei[E + e];
    float nrm = dinv[s] * dinv[d];
    float4 hv = *(const float4*)(h + s * GCN_D + f);
    float* __restrict__ o = agg + d * GCN_D + f;
    atomicAdd(o + 0, hv.x * nrm);
    atomicAdd(o + 1, hv.y * nrm);
    atomicAdd(o + 2, hv.z * nrm);
    atomicAdd(o + 3, hv.w * nrm);
}

// ---------------------------------------------------------------------------
// Epilogues.  Elementwise, so nxt may alias cur safely.
// ---------------------------------------------------------------------------
__global__ void k_post_relu_res(const float* __restrict__ agg,
                                const float* __restrict__ bias,
                                const float* __restrict__ cur,
                                float* __restrict__ nxt, int n64) {
    int i = blockIdx.x * blockDim.x + threadIdx.x;
    if (i < n64) {
        float v = agg[i] + bias[i & 63];
        v = v > 0.0f ? v : 0.0f;
        nxt[i] = v + cur[i];
    }
}

__global__ void k_post_bias(const float* __restrict__ agg,
                            const float* __restrict__ bias,
                            float* __restrict__ out, int n64) {
    int i = blockIdx.x * blockDim.x + threadIdx.x;
    if (i < n64) out[i] = agg[i] + bias[i & 63];
}

// ---------------------------------------------------------------------------
extern "C" void kernel_launch(void* const* d_in, const int* in_sizes, int n_in,
                              void* d_out, int out_size, void* d_ws, size_t ws_size,
                              hipStream_t stream) {
    const float*     x  = (const float*)d_in[0];
    const long long* ei = (const long long*)d_in[1];   // int64 per reference
    const float* W1 = (const float*)d_in[2];
    const float* b1 = (const float*)d_in[3];
    const float* W2 = (const float*)d_in[4];
    const float* b2 = (const float*)d_in[5];
    const float* W3 = (const float*)d_in[6];
    const float* b3 = (const float*)d_in[7];
    float* out = (float*)d_out;

    float* ws   = (float*)d_ws;
    float* dinv = ws;                          // N floats (deg, then dinv in place)
    float* h    = dinv + GCN_N;                // N*64
    float* agg  = h    + GCN_N * GCN_D;        // N*64
    float* buf  = agg  + GCN_N * GCN_D;        // N*64 ping buffer (layer outputs)

    const int n64    = GCN_N * GCN_D;          // 3.2M elements
    const int nQuads = n64 / 4;                // 800k float4
    const int TPB    = 256;                    // 8 waves per block (wave32)

    const int gN     = (GCN_N + TPB - 1) / TPB;
    const int gE     = (GCN_E + TPB - 1) / TPB;
    const int gQuads = (nQuads + TPB - 1) / TPB;
    const int gScat  = (GCN_E * 16 + TPB - 1) / TPB;
    const int gElem  = (n64 + TPB - 1) / TPB;
    const int nRowTiles = GCN_N / 16;          // 3125, exact
    const int gGemm  = (nRowTiles + (TPB / 32) - 1) / (TPB / 32);

    // --- normalization ------------------------------------------------------
    k_zero_f32   <<<gN, TPB, 0, stream>>>(dinv, GCN_N);
    k_count_deg  <<<gE, TPB, 0, stream>>>(ei, dinv, GCN_E);
    k_deg_to_dinv<<<gN, TPB, 0, stream>>>(dinv, GCN_N);

    // --- layer 0: conv -> relu -> residual(x) ------------------------------
    k_gemm64_wmma <<<gGemm,  TPB, 0, stream>>>(x, W1, h, nRowTiles);
    k_self_init   <<<gQuads, TPB, 0, stream>>>(h, dinv, agg, nQuads);
    k_edge_scatter<<<gScat,  TPB, 0, stream>>>(ei, h, dinv, agg, GCN_E);
    k_post_relu_res<<<gElem, TPB, 0, stream>>>(agg, b1, x, buf, n64);

    // --- layer 1: conv -> relu -> residual(buf), in-place ------------------
    k_gemm64_wmma <<<gGemm,  TPB, 0, stream>>>(buf, W2, h, nRowTiles);
    k_self_init   <<<gQuads, TPB, 0, stream>>>(h, dinv, agg, nQuads);
    k_edge_scatter<<<gScat,  TPB, 0, stream>>>(ei, h, dinv, agg, GCN_E);
    k_post_relu_res<<<gElem, TPB, 0, stream>>>(agg, b2, buf, buf, n64);

    // --- layer 2: conv only -------------------------------------------------
    k_gemm64_wmma <<<gGemm,  TPB, 0, stream>>>(buf, W3, h, nRowTiles);
    k_self_init   <<<gQuads, TPB, 0, stream>>>(h, dinv, agg, nQuads);
    k_edge_scatter<<<gScat,  TPB, 0, stream>>>(ei, h, dinv, agg, GCN_E);
    k_post_bias   <<<gElem,  TPB, 0, stream>>>(agg, b3, out, n64);
}